// LMMDLoss_18425409700337
// MI455X (gfx1250) — compile-verified
//
#include <hip/hip_runtime.h>
#include <hip/hip_bf16.h>
#include <math.h>
#include <stdint.h>

typedef float v2f __attribute__((ext_vector_type(2)));
typedef float v8f __attribute__((ext_vector_type(8)));
typedef unsigned int u32x4 __attribute__((ext_vector_type(4)));
typedef int i32x4 __attribute__((ext_vector_type(4)));
typedef int i32x8 __attribute__((ext_vector_type(8)));

#define BATCH   4096
#define DIM     512
#define NCLS    31
#define NTOT    8192      // 2*BATCH
#define TILE    128
#define KBLK    32
#define LDSP    36        // row pitch in floats: gcd(36,64)=4 -> conflict-free

#if __has_builtin(__builtin_amdgcn_tensor_load_to_lds) && __has_builtin(__builtin_amdgcn_s_wait_tensorcnt)
#  define USE_TDM 1
#  define NBUF 2
#else
#  define USE_TDM 0
#  define NBUF 1
#endif

// ---------------------------------------------------------------- small prep
__global__ void lmmd_zero(float* p, int n) {
    int i = blockIdx.x * 256 + threadIdx.x;
    if (i < n) p[i] = 0.0f;
}

__global__ void lmmd_row_sq(const float* __restrict__ src, const float* __restrict__ tgt,
                            float* __restrict__ sq, float* __restrict__ sumsq) {
    __shared__ float sh[256];
    int row = blockIdx.x;
    const float* p = (row < BATCH) ? (src + (size_t)row * DIM)
                                   : (tgt + (size_t)(row - BATCH) * DIM);
    float s = 0.0f;
    for (int c = threadIdx.x; c < DIM; c += 256) { float x = p[c]; s = fmaf(x, x, s); }
    sh[threadIdx.x] = s;  __syncthreads();
    for (int o = 128; o; o >>= 1) { if (threadIdx.x < o) sh[threadIdx.x] += sh[threadIdx.x + o]; __syncthreads(); }
    if (threadIdx.x == 0) { sq[row] = sh[0]; atomicAdd(sumsq, sh[0]); }
}

__global__ void lmmd_col_sum(const float* __restrict__ src, const float* __restrict__ tgt,
                             float* __restrict__ colsum) {
    int d = blockIdx.x * 256 + threadIdx.x;
    if (d >= DIM) return;
    float s = 0.0f;
    for (int r = 0; r < BATCH; ++r) s += src[(size_t)r * DIM + d];
    for (int r = 0; r < BATCH; ++r) s += tgt[(size_t)r * DIM + d];
    colsum[d] = s;
}

__global__ void lmmd_class_stats(const int* __restrict__ slab, const float* __restrict__ tl,
                                 int* __restrict__ cnt_s, int* __restrict__ present_t,
                                 float* __restrict__ t_colsum) {
    int r = blockIdx.x * 256 + threadIdx.x;
    if (r >= BATCH) return;
    atomicAdd(&cnt_s[slab[r]], 1);
    const float* row = tl + (size_t)r * NCLS;
    int best = 0; float bv = row[0];
    for (int c = 1; c < NCLS; ++c) { float v = row[c]; if (v > bv) { bv = v; best = c; } }
    atomicAdd(&present_t[best], 1);
    for (int c = 0; c < NCLS; ++c) atomicAdd(&t_colsum[c], row[c]);
}

__global__ void lmmd_finalize(const int* __restrict__ cnt_s, const int* __restrict__ present_t,
                              const float* __restrict__ t_colsum, const float* __restrict__ sumsq,
                              const float* __restrict__ colsum, const int* __restrict__ iter,
                              float* __restrict__ coefA, float* __restrict__ coefB,
                              float* __restrict__ scal) {
    __shared__ float shm[256];
    __shared__ float s_count;
    int t = threadIdx.x;
    if (t == 0) {
        float cnt = 0.0f;
        for (int c = 0; c < NCLS; ++c) cnt += (cnt_s[c] > 0 && present_t[c] > 0) ? 1.0f : 0.0f;
        s_count = cnt;
    }
    __syncthreads();
    float scale = (s_count > 0.0f) ? 1.0f / fmaxf(s_count, 1.0f) : 0.0f;
    if (t < 32) {
        float m    = (t < NCLS && cnt_s[t] > 0 && present_t[t] > 0) ? 1.0f : 0.0f;
        float coef = sqrtf(m * scale);
        float sdiv = (t < NCLS && cnt_s[t] > 0) ? (float)cnt_s[t] : 100.0f;
        float tsum = (t < NCLS) ? t_colsum[t] : 0.0f;
        float tdiv = (tsum == 0.0f) ? 100.0f : tsum;
        coefA[t] = (t < NCLS) ?  coef / sdiv : 0.0f;
        coefB[t] = (t < NCLS) ? -coef / tdiv : 0.0f;
    }
    float s = 0.0f;
    for (int d = t; d < DIM; d += 256) { float v = colsum[d]; s = fmaf(v, v, s); }
    shm[t] = s;  __syncthreads();
    for (int o = 128; o; o >>= 1) { if (t < o) shm[t] += shm[t + o]; __syncthreads(); }
    if (t == 0) {
        // sum(L2) = 2n*sum||x||^2 - 2*||sum x||^2  (diagonal clamp ~ 0)
        float sumL2 = 2.0f * (float)NTOT * sumsq[0] - 2.0f * shm[0];
        float bw = sumL2 / (float)((double)NTOT * (double)NTOT - (double)NTOT);
        bw = bw * 0.25f;                       // / KERNEL_MUL^(KERNEL_NUM//2)
        bw = fmaxf(bw, 1e-30f);
        float inv = 1.0f / bw;
        scal[0] = inv; scal[1] = inv * 0.5f; scal[2] = inv * 0.25f;
        scal[3] = inv * 0.125f; scal[4] = inv * 0.0625f;
        float p = (float)iter[0] / 1000.0f;
        scal[5] = 2.0f / (1.0f + __expf(-p)) - 1.0f;   // lamb, GAMMA=1
    }
}

__global__ void lmmd_build_V(const int* __restrict__ slab, const float* __restrict__ tl,
                             const float* __restrict__ coefA, const float* __restrict__ coefB,
                             float* __restrict__ V) {
    int idx = blockIdx.x * 256 + threadIdx.x;
    if (idx >= NTOT * 32) return;
    int row = idx >> 5, c = idx & 31;
    float v;
    if (row < BATCH) v = (c == slab[row]) ? coefA[c] : 0.0f;
    else             v = (c < NCLS) ? tl[(size_t)(row - BATCH) * NCLS + c] * coefB[c] : 0.0f;
    V[idx] = v;
}

// ---------------------------------------------------------------- TDM helper
#if USE_TDM
// 2D TDM descriptor: 128 x 32 f32 tile from a row-major [*,512] tensor into LDS,
// LDS padding 4 DWORDs after every 32 DWORDs -> 36-float pitch (bank-conflict-free).
__device__ static inline void tdm_load_panel(const float* gtile, float* lds) {
    uint64_t ga = (uint64_t)(uintptr_t)gtile;
    uint32_t lo = (uint32_t)(uintptr_t)lds;     // low 32 bits = LDS byte offset
    u32x4 g0;
    g0.x = 1u;                                           // count=1, user mode, no gather
    g0.y = lo;                                           // lds_addr
    g0.z = (uint32_t)ga;                                 // global_addr[31:0]
    g0.w = (uint32_t)((ga >> 32) & 0x1FFFFFFu)           // global_addr[56:32]
         | (2u << 30);                                   // type=2 ("image")
    i32x8 g1;
    g1[0] = (2 << 16)        // data_size = 4B
          | (1 << 20)        // pad_enable
          | (4 << 22)        // pad_interval: 32 DWORDs
          | (3 << 25);       // pad_amount:   4 DWORDs
    g1[1] = (int)(512u << 16);                 // tensor_dim0 = 512 (bits 79:48)
    g1[2] = (int)(8192u << 16);                // tensor_dim0 hi=0 | tensor_dim1 lo = 8192
    g1[3] = (int)(32u << 16);                  // tensor_dim1 hi=0 | tile_dim0 = 32
    g1[4] = 128;                               // tile_dim1 = 128, tile_dim2 = 0
    g1[5] = 512;                               // tensor_dim0_stride = 512
    g1[6] = 0;                                 // stride hi | tensor_dim1_stride lo
    g1[7] = 0;
    i32x4 z4 = {0, 0, 0, 0};
#if defined(__clang_major__) && __clang_major__ >= 23
    i32x8 z8 = {0, 0, 0, 0, 0, 0, 0, 0};
    __builtin_amdgcn_tensor_load_to_lds(g0, g1, z4, z4, z8, 0);
#else
    __builtin_amdgcn_tensor_load_to_lds(g0, g1, z4, z4, 0);
#endif
}
#endif

// ---------------------------------------------------------------- main fused kernel
__global__ __launch_bounds__(256)
void lmmd_main(const float* __restrict__ src, const float* __restrict__ tgt,
               const float* __restrict__ V, const float* __restrict__ sq,
               const float* __restrict__ scal, float* __restrict__ loss_acc) {
    __shared__ float As[NBUF][TILE * LDSP];
    __shared__ float Bs[NBUF][TILE * LDSP];
    __shared__ float red[8];

    const int tid  = threadIdx.x;
    const int wave = tid >> 5;
    const int lane = tid & 31;
    const int half = lane >> 4;       // 0 / 1 -> K offset {0,2}
    const int l15  = lane & 15;
    const int wm   = wave >> 1;       // 0..3 -> 32-row strip
    const int wn   = wave & 1;        // 0..1 -> 64-col strip

    const int Ibase = blockIdx.x * TILE;
    const int Jbase = blockIdx.y * TILE;
    const float* rowPtrI = (Ibase < BATCH) ? (src + (size_t)Ibase * DIM)
                                           : (tgt + (size_t)(Ibase - BATCH) * DIM);
    const float* rowPtrJ = (Jbase < BATCH) ? (src + (size_t)Jbase * DIM)
                                           : (tgt + (size_t)(Jbase - BATCH) * DIM);

    v8f acc[2][4] = {};

#if USE_TDM
    // prologue: wave0 DMAs the A panel, wave1 the B panel (panel 0 -> buffer 0)
    if (wave < 2) {
        const float* base = (wave == 0) ? rowPtrI : rowPtrJ;
        float* lbuf = (wave == 0) ? &As[0][0] : &Bs[0][0];
        tdm_load_panel(base, lbuf);
    }
#endif

    for (int kb = 0; kb < DIM / KBLK; ++kb) {
        const int k0 = kb * KBLK;
        __syncthreads();   // previous compute done -> safe to overwrite next buffer
#if USE_TDM
        if (wave < 2) {
            if (kb < DIM / KBLK - 1) {
                const float* base = (wave == 0) ? rowPtrI : rowPtrJ;
                float* lbuf = (wave == 0) ? &As[(kb + 1) & 1][0] : &Bs[(kb + 1) & 1][0];
                tdm_load_panel(base + k0 + KBLK, lbuf);        // issue next panel
                __builtin_amdgcn_s_wait_tensorcnt(1);          // current panel landed
            } else {
                __builtin_amdgcn_s_wait_tensorcnt(0);
            }
        }
#else
        // manual staging fallback: 128x32 panels of A and B (float4, pitch 36)
        #pragma unroll
        for (int i = 0; i < 4; ++i) {
            int idx = tid + i * 256;            // 0..1023
            int row = idx >> 3;
            int cv  = idx & 7;
            float4 a = *reinterpret_cast<const float4*>(rowPtrI + (size_t)row * DIM + k0 + cv * 4);
            float4 b = *reinterpret_cast<const float4*>(rowPtrJ + (size_t)row * DIM + k0 + cv * 4);
            *reinterpret_cast<float4*>(&As[0][row * LDSP + cv * 4]) = a;
            *reinterpret_cast<float4*>(&Bs[0][row * LDSP + cv * 4]) = b;
            if (k0 + KBLK < DIM) {
                __builtin_prefetch(rowPtrI + (size_t)row * DIM + k0 + KBLK + cv * 4, 0, 1);
                __builtin_prefetch(rowPtrJ + (size_t)row * DIM + k0 + KBLK + cv * 4, 0, 1);
            }
        }
#endif
        __syncthreads();

        const float* Ap = &As[kb % NBUF][0];
        const float* Bp = &Bs[kb % NBUF][0];

        #pragma unroll
        for (int kk = 0; kk < KBLK; kk += 4) {
            const int ck = kk + 2 * half;
            v2f a0 = *reinterpret_cast<const v2f*>(&Ap[(wm * 32 +  0 + l15) * LDSP + ck]);
            v2f a1 = *reinterpret_cast<const v2f*>(&Ap[(wm * 32 + 16 + l15) * LDSP + ck]);
            v2f b0 = *reinterpret_cast<const v2f*>(&Bp[(wn * 64 +  0 + l15) * LDSP + ck]);
            v2f b1 = *reinterpret_cast<const v2f*>(&Bp[(wn * 64 + 16 + l15) * LDSP + ck]);
            v2f b2 = *reinterpret_cast<const v2f*>(&Bp[(wn * 64 + 32 + l15) * LDSP + ck]);
            v2f b3 = *reinterpret_cast<const v2f*>(&Bp[(wn * 64 + 48 + l15) * LDSP + ck]);
            acc[0][0] = __builtin_amdgcn_wmma_f32_16x16x4_f32(false, a0, false, b0, (short)0, acc[0][0], false, false);
            acc[0][1] = __builtin_amdgcn_wmma_f32_16x16x4_f32(false, a0, false, b1, (short)0, acc[0][1], false, false);
            acc[0][2] = __builtin_amdgcn_wmma_f32_16x16x4_f32(false, a0, false, b2, (short)0, acc[0][2], false, false);
            acc[0][3] = __builtin_amdgcn_wmma_f32_16x16x4_f32(false, a0, false, b3, (short)0, acc[0][3], false, false);
            acc[1][0] = __builtin_amdgcn_wmma_f32_16x16x4_f32(false, a1, false, b0, (short)0, acc[1][0], false, false);
            acc[1][1] = __builtin_amdgcn_wmma_f32_16x16x4_f32(false, a1, false, b1, (short)0, acc[1][1], false, false);
            acc[1][2] = __builtin_amdgcn_wmma_f32_16x16x4_f32(false, a1, false, b2, (short)0, acc[1][2], false, false);
            acc[1][3] = __builtin_amdgcn_wmma_f32_16x16x4_f32(false, a1, false, b3, (short)0, acc[1][3], false, false);
        }
    }

    // epilogue: rank-31 weight tile via 8 more f32 WMMAs, then fused exp/reduce
    const float inv0 = scal[0], inv1 = scal[1], inv2 = scal[2], inv3 = scal[3], inv4 = scal[4];
    float partial = 0.0f;

    #pragma unroll
    for (int tm = 0; tm < 2; ++tm) {
        const int rI = Ibase + wm * 32 + tm * 16 + l15;
        #pragma unroll
        for (int tn = 0; tn < 4; ++tn) {
            const int rJ = Jbase + wn * 64 + tn * 16 + l15;
            v8f w = {};
            #pragma unroll
            for (int kk = 0; kk < 32; kk += 4) {
                v2f va = *reinterpret_cast<const v2f*>(&V[(size_t)rI * 32 + kk + 2 * half]);
                v2f vb = *reinterpret_cast<const v2f*>(&V[(size_t)rJ * 32 + kk + 2 * half]);
                w = __builtin_amdgcn_wmma_f32_16x16x4_f32(false, va, false, vb, (short)0, w, false, false);
            }
            const int rowb = Ibase + wm * 32 + tm * 16 + 8 * half;
            const int col  = Jbase + wn * 64 + tn * 16 + l15;
            const float sqc = sq[col];
            #pragma unroll
            for (int v = 0; v < 8; ++v) {
                float g  = acc[tm][tn][v];
                float l2 = fmaxf(sq[rowb + v] + sqc - 2.0f * g, 0.0f);
                float kern = __expf(-l2 * inv0) + __expf(-l2 * inv1) + __expf(-l2 * inv2)
                           + __expf(-l2 * inv3) + __expf(-l2 * inv4);
                partial = fmaf(w[v], kern, partial);
            }
        }
    }

    // wave32 shuffle reduce, then block reduce, one atomic per block
    for (int off = 16; off; off >>= 1) partial += __shfl_down(partial, off, 32);
    if (lane == 0) red[wave] = partial;
    __syncthreads();
    if (tid == 0) {
        float s = 0.0f;
        #pragma unroll
        for (int i = 0; i < 8; ++i) s += red[i];
        atomicAdd(loss_acc, s);
    }
}

__global__ void lmmd_write_out(const float* __restrict__ loss_acc,
                               const float* __restrict__ scal, float* __restrict__ out) {
    if (threadIdx.x == 0 && blockIdx.x == 0) out[0] = loss_acc[0] * scal[5];
}

// ---------------------------------------------------------------- launch
extern "C" void kernel_launch(void* const* d_in, const int* in_sizes, int n_in,
                              void* d_out, int out_size, void* d_ws, size_t ws_size,
                              hipStream_t stream) {
    const float* src  = (const float*)d_in[0];
    const float* tgt  = (const float*)d_in[1];
    const int*   slab = (const int*)d_in[2];
    const float* tl   = (const float*)d_in[3];
    const int*   iter = (const int*)d_in[4];
    float* out = (float*)d_out;

    char* ws = (char*)d_ws;
    float* V        = (float*)(ws);                         // 8192*32 f = 1 MiB
    float* sq       = (float*)(ws + 1048576);               // 8192 f
    float* colsum   = (float*)(ws + 1048576 + 32768);       // 512 f
    float* t_colsum = (float*)(ws + 1083392);               // 32 f
    int*   cnt_s    = (int*)  (ws + 1083520);               // 32 i
    int*   pres_t   = (int*)  (ws + 1083648);               // 32 i
    float* coefA    = (float*)(ws + 1083776);               // 32 f
    float* coefB    = (float*)(ws + 1083904);               // 32 f
    float* scal     = (float*)(ws + 1084032);               // 8 f (inv_denoms, lamb)
    float* sumsq    = (float*)(ws + 1084064);               // 1 f
    float* loss_acc = (float*)(ws + 1084068);               // 1 f

    // zero the accumulated stats region (t_colsum..loss_acc, contiguous)
    lmmd_zero<<<1, 256, 0, stream>>>(t_colsum, (int)((1084072 - 1083392) / 4));

    lmmd_row_sq     <<<NTOT, 256, 0, stream>>>(src, tgt, sq, sumsq);
    lmmd_col_sum    <<<2,    256, 0, stream>>>(src, tgt, colsum);
    lmmd_class_stats<<<BATCH/256, 256, 0, stream>>>(slab, tl, cnt_s, pres_t, t_colsum);
    lmmd_finalize   <<<1,    256, 0, stream>>>(cnt_s, pres_t, t_colsum, sumsq, colsum, iter,
                                               coefA, coefB, scal);
    lmmd_build_V    <<<(NTOT*32)/256, 256, 0, stream>>>(slab, tl, coefA, coefB, V);

    dim3 grid(NTOT / TILE, NTOT / TILE, 1);   // 64 x 64
    lmmd_main<<<grid, 256, 0, stream>>>(src, tgt, V, sq, scal, loss_acc);

    lmmd_write_out<<<1, 64, 0, stream>>>(loss_acc, scal, out);
}